// GeneralConvMaxAttention_58385785422516
// MI455X (gfx1250) — compile-verified
//
#include <hip/hip_runtime.h>
#include <hip/hip_bf16.h>
#include <stdint.h>

// ---------------------------------------------------------------------------
// GeneralConv (attention + max aggregation) for MI455X / gfx1250.
//
// Roofline: hoisting the per-edge linear layer to per-node (h = x @ W_msg)
// cuts FLOPs 16x (0.41 GFLOP GEMM). Remaining work is ~600MB of edge
// gathers/atomics -> HBM-bound (~30us @ 23.3 TB/s). The GEMM uses the f32
// WMMA path (v_wmma_f32_16x16x4_f32) so there is no precision change vs the
// f32 reference.
// ---------------------------------------------------------------------------

typedef __attribute__((ext_vector_type(2))) float v2f;
typedef __attribute__((ext_vector_type(8))) float v8f;

#define LDS_STRIDE 68          // 64 + pad: conflict-free b64 LDS reads
#define NEG_SLOPE 0.2f
#define NEG_INF_BITS 0xFF800000u

// IEEE float atomic max via sign-split integer atomics (memory init = -inf).
__device__ __forceinline__ void atomicMaxF32(float* addr, float val) {
    int iv = __float_as_int(val);
    if (iv >= 0) {
        atomicMax((int*)addr, iv);                 // positive: signed int max
    } else {
        atomicMin((unsigned int*)addr, (unsigned int)iv);  // negative: uint min
    }
}

// ---------------------------------------------------------------------------
// Kernel 0: initialize accumulators. out = -inf (segment-max identity),
// m = -inf, denom = 0.
// ---------------------------------------------------------------------------
__global__ __launch_bounds__(256) void k_init(float* __restrict__ out,
                                              float* __restrict__ mbuf,
                                              float* __restrict__ denom,
                                              int N) {
    int idx = blockIdx.x * 256 + threadIdx.x;
    const float ninf = __int_as_float((int)NEG_INF_BITS);
    if (idx < N * 64) out[idx] = ninf;
    if (idx < N) { mbuf[idx] = ninf; denom[idx] = 0.0f; }
}

// ---------------------------------------------------------------------------
// Kernel 1: hb[N,64] = x @ W_msg + (b_msg + b_edge), via f32 WMMA.
// One wave computes a 16x64 output tile (4 N-tiles of 16x16), K consumed
// 4 at a time: 16 steps x 4 tiles = 64 v_wmma_f32_16x16x4_f32 per wave.
// W is staged transposed in LDS so each B fragment is one aligned b64 read.
// ---------------------------------------------------------------------------
__global__ __launch_bounds__(256) void k_gemm_hb(const float* __restrict__ x,
                                                 const float* __restrict__ W,
                                                 const float* __restrict__ b_msg,
                                                 const float* __restrict__ b_edge,
                                                 float* __restrict__ hb,
                                                 int N) {
    __shared__ __align__(16) float ldsWT[64 * LDS_STRIDE];
    const int tid = threadIdx.x;

    // Stage W transposed: ldsWT[n*LDS_STRIDE + k] = W[k*64 + n]
    for (int idx = tid; idx < 64 * 64; idx += 256) {
        int k = idx >> 6, n = idx & 63;
        ldsWT[n * LDS_STRIDE + k] = W[idx];
    }
    __syncthreads();

    const int wave = tid >> 5;                 // 8 waves / block
    const int lane = tid & 31;
    const int row_tile = blockIdx.x * 8 + wave;
    const int m0 = row_tile * 16;
    if (m0 >= N) return;                        // no further barriers

    const int mlane = lane & 15;                // M (for A) / N (for B,C,D)
    const int khalf = (lane >> 4) * 2;          // lanes 0-15: K+0,1; 16-31: K+2,3

    v8f acc0 = {}, acc1 = {}, acc2 = {}, acc3 = {};
    const float* xrow = x + (size_t)(m0 + mlane) * 64;

#pragma unroll
    for (int k = 0; k < 64; k += 4) {
        v2f a  = *(const v2f*)(xrow + k + khalf);
        v2f b0 = *(const v2f*)(&ldsWT[( 0 + mlane) * LDS_STRIDE + k + khalf]);
        v2f b1 = *(const v2f*)(&ldsWT[(16 + mlane) * LDS_STRIDE + k + khalf]);
        v2f b2 = *(const v2f*)(&ldsWT[(32 + mlane) * LDS_STRIDE + k + khalf]);
        v2f b3 = *(const v2f*)(&ldsWT[(48 + mlane) * LDS_STRIDE + k + khalf]);
        acc0 = __builtin_amdgcn_wmma_f32_16x16x4_f32(false, a, false, b0, (short)0, acc0, false, false);
        acc1 = __builtin_amdgcn_wmma_f32_16x16x4_f32(false, a, false, b1, (short)0, acc1, false, false);
        acc2 = __builtin_amdgcn_wmma_f32_16x16x4_f32(false, a, false, b2, (short)0, acc2, false, false);
        acc3 = __builtin_amdgcn_wmma_f32_16x16x4_f32(false, a, false, b3, (short)0, acc3, false, false);
    }

    // Bias is per-column: n = t*16 + mlane
    const float bias0 = b_msg[ 0 + mlane] + b_edge[ 0 + mlane];
    const float bias1 = b_msg[16 + mlane] + b_edge[16 + mlane];
    const float bias2 = b_msg[32 + mlane] + b_edge[32 + mlane];
    const float bias3 = b_msg[48 + mlane] + b_edge[48 + mlane];

    const int mrow = (lane >> 4) * 8;           // C/D: VGPR v -> M = v + 8*(l/16)
#pragma unroll
    for (int v = 0; v < 8; ++v) {
        float* orow = hb + (size_t)(m0 + mrow + v) * 64;
        orow[ 0 + mlane] = acc0[v] + bias0;
        orow[16 + mlane] = acc1[v] + bias1;
        orow[32 + mlane] = acc2[v] + bias2;
        orow[48 + mlane] = acc3[v] + bias3;
    }
}

// ---------------------------------------------------------------------------
// Kernel 2: per-edge attention logit + segment max into m[dst].
// One wave per edge: lane covers features {lane, lane+32} (coalesced 128B
// gathers of hb[src]); butterfly reduce across the wave32.
// ---------------------------------------------------------------------------
__global__ __launch_bounds__(256) void k_edge_logits(const float* __restrict__ hb,
                                                     const long long* __restrict__ ei,
                                                     const float* __restrict__ edge_attr,
                                                     const float* __restrict__ W_edge,
                                                     const float* __restrict__ att,
                                                     float* __restrict__ logits,
                                                     float* __restrict__ mbuf,
                                                     int E) {
    const int lane = threadIdx.x & 31;
    const int e = blockIdx.x * 8 + (threadIdx.x >> 5);
    if (e >= E) return;

    const long long s = ei[e];
    const long long d = ei[(size_t)E + e];
    const float ea = edge_attr[e];
    const float* hrow = hb + (size_t)s * 64;

    float sum = 0.0f;
#pragma unroll
    for (int i = 0; i < 2; ++i) {
        const int dd = lane + i * 32;
        float v = hrow[dd] + ea * W_edge[dd];
        v = v > 0.0f ? v : NEG_SLOPE * v;       // leaky_relu(msg)
        sum += v * att[dd];
    }
#pragma unroll
    for (int off = 16; off > 0; off >>= 1) sum += __shfl_xor(sum, off, 32);

    if (lane == 0) {
        logits[e] = sum;
        atomicMaxF32(&mbuf[(size_t)d], sum);
    }
}

// ---------------------------------------------------------------------------
// Kernel 3: ex = exp(logit - m[dst]); denom[dst] += ex. (thread per edge)
// ---------------------------------------------------------------------------
__global__ __launch_bounds__(256) void k_edge_exp(const long long* __restrict__ ei,
                                                  float* __restrict__ logits,   // in: logit, out: ex
                                                  const float* __restrict__ mbuf,
                                                  float* __restrict__ denom,
                                                  int E) {
    const int e = blockIdx.x * 256 + threadIdx.x;
    if (e >= E) return;
    const long long d = ei[(size_t)E + e];
    const float ex = __expf(logits[e] - mbuf[(size_t)d]);
    logits[e] = ex;
    atomicAdd(&denom[(size_t)d], ex);
}

// ---------------------------------------------------------------------------
// Kernel 4: weighted = msg * alpha; per-feature segment MAX into out[dst].
// Wave per edge; recomputes msg from hb[src] (cheaper than storing E x D).
// ---------------------------------------------------------------------------
__global__ __launch_bounds__(256) void k_edge_aggregate(const float* __restrict__ hb,
                                                        const long long* __restrict__ ei,
                                                        const float* __restrict__ edge_attr,
                                                        const float* __restrict__ W_edge,
                                                        const float* __restrict__ exbuf,
                                                        const float* __restrict__ denom,
                                                        float* __restrict__ out,
                                                        int E) {
    const int lane = threadIdx.x & 31;
    const int e = blockIdx.x * 8 + (threadIdx.x >> 5);
    if (e >= E) return;

    const long long s = ei[e];
    const long long d = ei[(size_t)E + e];
    const float ea = edge_attr[e];
    const float alpha = exbuf[e] / (denom[(size_t)d] + 1e-16f);

    const float* hrow = hb + (size_t)s * 64;
    float* orow = out + (size_t)d * 64;
#pragma unroll
    for (int i = 0; i < 2; ++i) {
        const int dd = lane + i * 32;
        const float w = (hrow[dd] + ea * W_edge[dd]) * alpha;
        atomicMaxF32(&orow[dd], w);
    }
}

// ---------------------------------------------------------------------------
// Kernel 5: out = (out == -inf ? 0 : out) + x   (residual / empty segments)
// ---------------------------------------------------------------------------
__global__ __launch_bounds__(256) void k_finalize(const float* __restrict__ x,
                                                  float* __restrict__ out,
                                                  int total) {
    const int idx = blockIdx.x * 256 + threadIdx.x;
    if (idx >= total) return;
    float v = out[idx];
    if (__float_as_uint(v) == NEG_INF_BITS) v = 0.0f;
    out[idx] = v + x[idx];
}

// ---------------------------------------------------------------------------
// Launch: inputs in setup_inputs() order:
//   0 x[N,64] f32, 1 edge_index[2,E] i64, 2 edge_attr[E] f32, 3 W_msg[64,64],
//   4 b_msg[64], 5 W_edge[1,64], 6 b_edge[64], 7 att[64]
// ---------------------------------------------------------------------------
extern "C" void kernel_launch(void* const* d_in, const int* in_sizes, int n_in,
                              void* d_out, int out_size, void* d_ws, size_t ws_size,
                              hipStream_t stream) {
    const float*     x         = (const float*)d_in[0];
    const long long* ei        = (const long long*)d_in[1];   // int64 per reference
    const float*     edge_attr = (const float*)d_in[2];
    const float*     W_msg     = (const float*)d_in[3];
    const float*     b_msg     = (const float*)d_in[4];
    const float*     W_edge    = (const float*)d_in[5];
    const float*     b_edge    = (const float*)d_in[6];
    const float*     att       = (const float*)d_in[7];
    float*           out       = (float*)d_out;

    const int N = in_sizes[0] / 64;
    const int E = in_sizes[2];

    // Workspace layout (floats): hb[N*64] | logits/ex[E] | m[N] | denom[N]
    float* hb     = (float*)d_ws;
    float* logits = hb + (size_t)N * 64;
    float* mbuf   = logits + E;
    float* denom  = mbuf + N;

    // 0) init accumulators
    {
        int total = N * 64;
        k_init<<<(total + 255) / 256, 256, 0, stream>>>(out, mbuf, denom, N);
    }
    // 1) hb = x @ W_msg + (b_msg + b_edge)   [WMMA f32]
    {
        int tiles = (N + 15) / 16;
        k_gemm_hb<<<(tiles + 7) / 8, 256, 0, stream>>>(x, W_msg, b_msg, b_edge, hb, N);
    }
    // 2) logits + segment max
    k_edge_logits<<<(E + 7) / 8, 256, 0, stream>>>(hb, ei, edge_attr, W_edge, att,
                                                   logits, mbuf, E);
    // 3) exp + segment sum
    k_edge_exp<<<(E + 255) / 256, 256, 0, stream>>>(ei, logits, mbuf, denom, E);
    // 4) weighted segment max aggregation
    k_edge_aggregate<<<(E + 7) / 8, 256, 0, stream>>>(hb, ei, edge_attr, W_edge,
                                                      logits, denom, out, E);
    // 5) residual + empty-segment fixup
    {
        int total = N * 64;
        k_finalize<<<(total + 255) / 256, 256, 0, stream>>>(x, out, total);
    }
}